// DrugGATEncoder_71751723647149
// MI455X (gfx1250) — compile-verified
//
#include <hip/hip_runtime.h>
#include <math.h>

// ---- problem constants (dims of the DrugGATEncoder reference) ----
#define D_FEAT    128      // IN == HEADS*HID == OUT == 128
#define HID_C     64
#define HEADS1    2
#define NEG_SLOPE 0.2f

typedef float v2f __attribute__((ext_vector_type(2)));
typedef float v8f __attribute__((ext_vector_type(8)));

// =====================================================================
// WMMA f32 GEMM:  C[n x 128] = A[n x 128] @ W[128 x 128]
// One block = 16 rows of A; 8 waves each own a 16x16 column tile.
// W panels staged via GLOBAL_LOAD_ASYNC_TO_LDS_B32 (ASYNCcnt path),
// MACs via V_WMMA_F32_16X16X4_F32 (wave32), K swept in steps of 4.
// =====================================================================
__global__ __launch_bounds__(256) void gemm_nk128(const float* __restrict__ A,
                                                  const float* __restrict__ W,
                                                  float* __restrict__ C, int n) {
    __shared__ float sA[16][132];   // padded: column reads hit distinct banks
    __shared__ float sW[32][128];

    const int tid = threadIdx.x;
    const int m0  = blockIdx.x * 16;

    // stage the 16x128 A tile (zero-padded at the tail of N) -- normal path,
    // the guard needs zero-fill which async-to-LDS can't give us.
    for (int i = tid; i < 16 * 128; i += 256) {
        int r = i >> 7, c = i & 127;
        int gm = m0 + r;
        sA[r][c] = (gm < n) ? A[(size_t)gm * D_FEAT + c] : 0.0f;
    }

    const unsigned ldsW0 = (unsigned)(size_t)(void*)&sW[0][0]; // LDS byte offset
    const int wave = tid >> 5;
    const int lane = tid & 31;
    const int half = lane >> 4;       // 0: K lo pair, 1: K hi pair (A layout)
    const int l15  = lane & 15;
    const int kk   = half * 2;
    const int colN = wave * 16 + l15; // this lane's N index in B / C

    v8f acc = {};
    for (int k0 = 0; k0 < 128; k0 += 32) {
        __syncthreads();
        // async cache->LDS stage of the 32x128 W panel (no VGPR data path)
        for (int i = tid; i < 32 * 128; i += 256) {
            unsigned     lds = ldsW0 + (unsigned)i * 4u;       // sW is contiguous
            const float* gp  = W + (size_t)k0 * D_FEAT + i;
            asm volatile("global_load_async_to_lds_b32 %0, %1, off"
                         :: "v"(lds), "v"(gp) : "memory");
        }
        asm volatile("s_wait_asynccnt 0x0" ::: "memory");       // my copies done
        __syncthreads();                                        // everyone's done

        #pragma unroll
        for (int k = 0; k < 32; k += 4) {
            v2f a, b;
            a.x = sA[l15][k0 + k + kk];
            a.y = sA[l15][k0 + k + kk + 1];
            b.x = sW[k + kk][colN];
            b.y = sW[k + kk + 1][colN];
            acc = __builtin_amdgcn_wmma_f32_16x16x4_f32(
                false, a, false, b, (short)0, acc, false, false);
        }
    }

    // C/D layout: VGPR r -> M = r (+8 for lanes 16..31), N = lane&15
    const int mBase = m0 + half * 8;
    #pragma unroll
    for (int r = 0; r < 8; ++r) {
        int m = mBase + r;
        if (m < n) C[(size_t)m * D_FEAT + colN] = acc[r];
    }
}

// =====================================================================
// attention logits: a_s[n,h] = <H[n,h,:], att_src[h,:]>, same for a_d
// =====================================================================
__global__ void attn_logits(const float* __restrict__ H,
                            const float* __restrict__ attS,
                            const float* __restrict__ attD,
                            float* __restrict__ a_s, float* __restrict__ a_d,
                            int n, int heads, int ch) {
    int idx = blockIdx.x * blockDim.x + threadIdx.x;
    if (idx >= n * heads) return;
    int node = idx / heads, hh = idx - node * heads;
    const float* hp = H + (size_t)node * D_FEAT + hh * ch;
    const float* sp = attS + hh * ch;
    const float* dp = attD + hh * ch;
    float vs = 0.f, vd = 0.f;
    for (int c = 0; c < ch; ++c) {
        float hv = hp[c];
        vs += hv * sp[c];
        vd += hv * dp[c];
    }
    a_s[idx] = vs;
    a_d[idx] = vd;
}

// ---- monotone uint key for float atomic-max over possibly-negative data ----
__device__ __forceinline__ unsigned fkey(float x) {
    unsigned u = __float_as_uint(x);
    return (u & 0x80000000u) ? ~u : (u | 0x80000000u);
}
__device__ __forceinline__ float funkey(unsigned k) {
    return (k & 0x80000000u) ? __uint_as_float(k & 0x7FFFFFFFu)
                             : __uint_as_float(~k);
}

__device__ __forceinline__ void edge_endpoints(const int* __restrict__ ei, int E,
                                               int e, int& src, int& dst) {
    if (e < E) { src = ei[e]; dst = ei[E + e]; }
    else       { src = dst = e - E; }   // appended self-loops
}

// pass 1: e = leaky_relu(a_s[src] + a_d[dst]); per-dst running max
__global__ void edge_pass1(const int* __restrict__ ei,
                           const float* __restrict__ a_s,
                           const float* __restrict__ a_d,
                           float* __restrict__ ebuf,
                           unsigned* __restrict__ mkey,
                           int E, int n, int heads) {
    int idx = blockIdx.x * blockDim.x + threadIdx.x;
    int etot = E + n;
    if (idx >= etot * heads) return;
    int e = idx / heads, hh = idx - e * heads;
    int src, dst; edge_endpoints(ei, E, e, src, dst);
    float v = a_s[src * heads + hh] + a_d[dst * heads + hh];
    v = (v > 0.f) ? v : NEG_SLOPE * v;
    ebuf[idx] = v;
    atomicMax(&mkey[dst * heads + hh], fkey(v));
}

// pass 2: p = exp(e - max[dst]); denom[dst] += p
__global__ void edge_pass2(const int* __restrict__ ei,
                           float* __restrict__ ebuf,
                           const unsigned* __restrict__ mkey,
                           float* __restrict__ denom,
                           int E, int n, int heads) {
    int idx = blockIdx.x * blockDim.x + threadIdx.x;
    int etot = E + n;
    if (idx >= etot * heads) return;
    int e = idx / heads, hh = idx - e * heads;
    int src, dst; edge_endpoints(ei, E, e, src, dst);
    (void)src;
    float p = expf(ebuf[idx] - funkey(mkey[dst * heads + hh]));
    ebuf[idx] = p;
    atomicAdd(&denom[dst * heads + hh], p);
}

// pass 3: alpha = p / (denom[dst] + eps)
__global__ void edge_alpha(const int* __restrict__ ei,
                           float* __restrict__ ebuf,
                           const float* __restrict__ denom,
                           int E, int n, int heads) {
    int idx = blockIdx.x * blockDim.x + threadIdx.x;
    int etot = E + n;
    if (idx >= etot * heads) return;
    int e = idx / heads, hh = idx - e * heads;
    int src, dst; edge_endpoints(ei, E, e, src, dst);
    (void)src;
    ebuf[idx] = ebuf[idx] / (denom[dst * heads + hh] + 1e-16f);
}

// pass 4: agg[dst, c] += alpha[e,h(c)] * H[src, c]   (128 lanes per edge)
__global__ void edge_aggregate(const int* __restrict__ ei,
                               const float* __restrict__ ebuf,
                               const float* __restrict__ H,
                               float* __restrict__ agg,
                               int E, int n, int heads, int ch) {
    int idx = blockIdx.x * blockDim.x + threadIdx.x;
    int etot = E + n;
    if (idx >= etot * D_FEAT) return;
    int e = idx >> 7, c = idx & 127;
    int src, dst; edge_endpoints(ei, E, e, src, dst);
    int hh = c / ch;                      // ch=64,heads=2 -> c>>6; ch=128 -> 0
    float alpha = ebuf[e * heads + hh];
    atomicAdd(&agg[(size_t)dst * D_FEAT + c], alpha * H[(size_t)src * D_FEAT + c]);
}

// out = agg + bias, optional ELU (alpha=1)
__global__ void bias_act(const float* __restrict__ agg,
                         const float* __restrict__ b,
                         float* __restrict__ out, int n, int do_elu) {
    int idx = blockIdx.x * blockDim.x + threadIdx.x;
    if (idx >= n * D_FEAT) return;
    float v = agg[idx] + b[idx & 127];
    if (do_elu) v = (v > 0.f) ? v : expm1f(v);
    out[idx] = v;
}

__global__ void zero_f(float* __restrict__ p, long long count) {
    long long i = (long long)blockIdx.x * blockDim.x + threadIdx.x;
    if (i < count) p[i] = 0.f;
}

// ---- global mean pool ----
__global__ void pool_count(const int* __restrict__ batch, float* __restrict__ counts, int n) {
    int i = blockIdx.x * blockDim.x + threadIdx.x;
    if (i < n) atomicAdd(&counts[batch[i]], 1.0f);
}
__global__ void pool_sum(const int* __restrict__ batch, const float* __restrict__ h,
                         float* __restrict__ out, int n) {
    int idx = blockIdx.x * blockDim.x + threadIdx.x;
    if (idx >= n * D_FEAT) return;
    int node = idx >> 7, c = idx & 127;
    atomicAdd(&out[(size_t)batch[node] * D_FEAT + c], h[idx]);
}
__global__ void pool_div(float* __restrict__ out, const float* __restrict__ counts, int g) {
    int idx = blockIdx.x * blockDim.x + threadIdx.x;
    if (idx >= g * D_FEAT) return;
    out[idx] /= fmaxf(counts[idx >> 7], 1.0f);
}

// =====================================================================
// host side
// =====================================================================
static inline int nblk(long long total) { return (int)((total + 255) / 256); }

static void run_gat_layer(const float* Xin, const float* W,
                          const float* attS, const float* attD, const float* bias,
                          int heads, int ch, int do_elu,
                          const int* ei, int E, int n,
                          float* Hbuf, float* aggbuf, float* ebuf,
                          float* a_s, float* a_d, unsigned* mkey, float* denom,
                          hipStream_t stream) {
    const int etot = E + n;
    gemm_nk128<<<(n + 15) / 16, 256, 0, stream>>>(Xin, W, Hbuf, n);
    attn_logits<<<nblk((long long)n * heads), 256, 0, stream>>>(Hbuf, attS, attD, a_s, a_d, n, heads, ch);
    zero_f<<<nblk((long long)n * heads), 256, 0, stream>>>((float*)mkey, (long long)n * heads);
    zero_f<<<nblk((long long)n * heads), 256, 0, stream>>>(denom, (long long)n * heads);
    zero_f<<<nblk((long long)n * D_FEAT), 256, 0, stream>>>(aggbuf, (long long)n * D_FEAT);
    edge_pass1<<<nblk((long long)etot * heads), 256, 0, stream>>>(ei, a_s, a_d, ebuf, mkey, E, n, heads);
    edge_pass2<<<nblk((long long)etot * heads), 256, 0, stream>>>(ei, ebuf, mkey, denom, E, n, heads);
    edge_alpha<<<nblk((long long)etot * heads), 256, 0, stream>>>(ei, ebuf, denom, E, n, heads);
    edge_aggregate<<<nblk((long long)etot * D_FEAT), 256, 0, stream>>>(ei, ebuf, Hbuf, aggbuf, E, n, heads, ch);
    bias_act<<<nblk((long long)n * D_FEAT), 256, 0, stream>>>(aggbuf, bias, aggbuf, n, do_elu);
}

extern "C" void kernel_launch(void* const* d_in, const int* in_sizes, int n_in,
                              void* d_out, int out_size, void* d_ws, size_t ws_size,
                              hipStream_t stream) {
    const float* x     = (const float*)d_in[0];
    const int*   ei    = (const int*)  d_in[1];
    const int*   batch = (const int*)  d_in[2];
    const float* W1    = (const float*)d_in[3];
    const float* as1   = (const float*)d_in[4];
    const float* ad1   = (const float*)d_in[5];
    const float* b1    = (const float*)d_in[6];
    const float* W2    = (const float*)d_in[7];
    const float* as2   = (const float*)d_in[8];
    const float* ad2   = (const float*)d_in[9];
    const float* b2    = (const float*)d_in[10];
    float* out = (float*)d_out;

    const int n = in_sizes[0] / D_FEAT;
    const int E = in_sizes[1] / 2;
    const int g = out_size / D_FEAT;
    const int etot = E + n;

    // workspace carve-up (floats)
    float* ws = (float*)d_ws;
    size_t off = 0;
    float*    bufA  = ws + off; off += (size_t)n * D_FEAT;     // H1pre -> H2pre
    float*    bufB  = ws + off; off += (size_t)n * D_FEAT;     // agg1/h1 -> agg2/out2
    float*    ebuf  = ws + off; off += (size_t)etot * HEADS1;  // e -> p -> alpha
    float*    a_s   = ws + off; off += (size_t)n * HEADS1;
    float*    a_d   = ws + off; off += (size_t)n * HEADS1;
    unsigned* mkey  = (unsigned*)(ws + off); off += (size_t)n * HEADS1;
    float*    denom = ws + off; off += (size_t)n * HEADS1;
    float*    cnts  = ws + off; off += (size_t)g;
    (void)ws_size; (void)n_in;

    // layer 1: GAT(128 -> 2x64, concat) + ELU      h1 -> bufB
    run_gat_layer(x, W1, as1, ad1, b1, HEADS1, HID_C, /*elu=*/1,
                  ei, E, n, bufA, bufB, ebuf, a_s, a_d, mkey, denom, stream);

    // layer 2: GAT(128 -> 1x128, mean==identity)   out2 -> bufB
    run_gat_layer(bufB, W2, as2, ad2, b2, /*heads=*/1, /*ch=*/D_FEAT, /*elu=*/0,
                  ei, E, n, bufA, bufB, ebuf, a_s, a_d, mkey, denom, stream);

    // global mean pool over graphs
    zero_f<<<nblk((long long)g * D_FEAT), 256, 0, stream>>>(out, (long long)g * D_FEAT);
    zero_f<<<nblk((long long)g), 256, 0, stream>>>(cnts, (long long)g);
    pool_count<<<nblk((long long)n), 256, 0, stream>>>(batch, cnts, n);
    pool_sum<<<nblk((long long)n * D_FEAT), 256, 0, stream>>>(batch, bufB, out, n);
    pool_div<<<nblk((long long)g * D_FEAT), 256, 0, stream>>>(out, cnts, g);
}